// SimplePhiEngine_50190987821652
// MI455X (gfx1250) — compile-verified
//
#include <hip/hip_runtime.h>
#include <cstdint>

// Langevin dynamics on MI455X (gfx1250):
//   x <- x - grad_V(x)*DT + SIGMA*N(0,1),  grad_V = 2*J2*x + 4*J4*x^3 + b
// Strategy: state lives in registers for all n_steps; exactly one HBM read of
// x0 and one write of x (4 MB total -> ~0.17us at 23.3 TB/s). Noise from a
// counter-based Philox4x32-10 keyed on (element-group, step) -> deterministic,
// replay-safe. Bias vector staged into LDS via the CDNA5 async-to-LDS path.

#define LJ2  (-1.0f)
#define LJ4  (0.5f)
#define LDT  (0.01f)
#define LSIGMA (0.1f)   // sqrt(2*KT*DT) = sqrt(2*0.5*0.01) = 0.1 exactly

__device__ __forceinline__ void philox_round(uint32_t& c0, uint32_t& c1,
                                             uint32_t& c2, uint32_t& c3,
                                             uint32_t k0, uint32_t k1) {
    const uint32_t M0 = 0xD2511F53u, M1 = 0xCD9E8D57u;
    uint32_t hi0 = __umulhi(M0, c0), lo0 = M0 * c0;
    uint32_t hi1 = __umulhi(M1, c2), lo1 = M1 * c2;
    uint32_t n0 = hi1 ^ c1 ^ k0;
    uint32_t n1 = lo1;
    uint32_t n2 = hi0 ^ c3 ^ k1;
    uint32_t n3 = lo0;
    c0 = n0; c1 = n1; c2 = n2; c3 = n3;
}

__device__ __forceinline__ void philox4x32_10(uint32_t tid, uint32_t step,
                                              uint32_t out[4]) {
    uint32_t c0 = tid, c1 = step, c2 = 0x85A308D3u, c3 = 0xC2B2AE35u;
    uint32_t k0 = 0xA511E9B3u, k1 = 0x9E3779B9u;
#pragma unroll
    for (int r = 0; r < 10; ++r) {
        philox_round(c0, c1, c2, c3, k0, k1);
        k0 += 0x9E3779B9u;
        k1 += 0xBB67AE85u;
    }
    out[0] = c0; out[1] = c1; out[2] = c2; out[3] = c3;
}

// Box-Muller: two u32 -> two standard normals (fast transcendental path).
__device__ __forceinline__ void box_muller(uint32_t a, uint32_t b,
                                           float& n0, float& n1) {
    // u1 in (0,1], u2 in [0,1)
    float u1 = fmaf((float)(a >> 8), 0x1p-24f, 0x1p-25f);
    float u2 = (float)(b >> 8) * 0x1p-24f;
    float r  = sqrtf(-2.0f * __logf(u1));
    float sn, cs;
    __sincosf(6.28318530717958647692f * u2, &sn, &cs);
    n0 = r * cs;
    n1 = r * sn;
}

// One Euler-Maruyama step for one element.
__device__ __forceinline__ float lstep(float x, float b, float n) {
    float x3   = x * x * x;
    float grad = fmaf(2.0f * LJ2, x, fmaf(4.0f * LJ4, x3, b));
    return fmaf(LSIGMA, n, fmaf(-LDT, grad, x));
}

__global__ __launch_bounds__(256)
void langevin500_kernel(const float* __restrict__ x0,
                        const float* __restrict__ bias,
                        const int*   __restrict__ n_steps_p,
                        float*       __restrict__ out,
                        int total, int n_dim) {
    __shared__ float s_b[256];

    // --- Stage bias into LDS via CDNA5 async global->LDS (ASYNCcnt path) ---
    if ((int)threadIdx.x < n_dim) {
        const float* gsrc = bias + threadIdx.x;
        typedef __attribute__((address_space(3))) float lds_f;
        lds_f* sp = (lds_f*)&s_b[threadIdx.x];
        uint32_t lds_off = (uint32_t)(unsigned long long)sp;
        asm volatile("global_load_async_to_lds_b32 %0, %1, off"
                     :: "v"(lds_off), "v"(gsrc)
                     : "memory");
    }
    asm volatile("s_wait_asynccnt 0" ::: "memory");
    __syncthreads();

    int tid  = blockIdx.x * blockDim.x + threadIdx.x;
    int base = tid * 4;
    if (base + 3 >= total) return;   // total is a multiple of 4 (2048*256)

    int nsteps = n_steps_p[0];       // device-side read: graph-capture safe

    // One 128-bit load for this thread's 4 elements.
    float4 x = *(const float4*)(x0 + base);

    int d0 = base & (n_dim - 1);     // n_dim is a power of two (256)
    float b0 = s_b[d0 + 0];
    float b1 = s_b[d0 + 1];
    float b2 = s_b[d0 + 2];
    float b3 = s_b[d0 + 3];

    for (int s = 0; s < nsteps; ++s) {
        uint32_t r[4];
        philox4x32_10((uint32_t)tid, (uint32_t)s, r);
        float n0, n1, n2, n3;
        box_muller(r[0], r[1], n0, n1);
        box_muller(r[2], r[3], n2, n3);
        x.x = lstep(x.x, b0, n0);
        x.y = lstep(x.y, b1, n1);
        x.z = lstep(x.z, b2, n2);
        x.w = lstep(x.w, b3, n3);
    }

    *(float4*)(out + base) = x;      // one 128-bit store
}

extern "C" void kernel_launch(void* const* d_in, const int* in_sizes, int n_in,
                              void* d_out, int out_size, void* d_ws, size_t ws_size,
                              hipStream_t stream) {
    const float* x0      = (const float*)d_in[0];   // (2048, 256) fp32
    const float* bias    = (const float*)d_in[1];   // (256,)      fp32
    const int*   n_steps = (const int*)d_in[2];     // scalar int on device

    float* out = (float*)d_out;

    int total = in_sizes[0];          // 524288
    int n_dim = in_sizes[1];          // 256
    int nvec  = total / 4;            // 4 elements per thread
    int block = 256;                  // 8 wave32 waves per workgroup
    int grid  = (nvec + block - 1) / block;

    langevin500_kernel<<<grid, block, 0, stream>>>(x0, bias, n_steps, out,
                                                   total, n_dim);
}